// FlowCalculator_64072322121841
// MI455X (gfx1250) — compile-verified
//
#include <hip/hip_runtime.h>

// Problem constants (match the reference).
#define BB   32
#define HH   512
#define WW   384
#define VV   6890
#define FF   13776
#define PADX 64
#define WPAD 512          // WW + 2*PADX
#define ROWS 8            // padded rows per block in the main kernel

// ---------------------------------------------------------------------------
// Kernel 1: zero the per-(batch,face) visibility byte array (440,832 bytes).
// ---------------------------------------------------------------------------
__global__ void vf2_zero_kernel(unsigned int* __restrict__ p, int n32) {
    int i = blockIdx.x * blockDim.x + threadIdx.x;
    if (i < n32) p[i] = 0u;
}

// ---------------------------------------------------------------------------
// Kernel 2: scatter visibility of faces seen in view 2.
// grid = (H*W/256, B). Races are benign (every store writes 1).
// ---------------------------------------------------------------------------
__global__ void vf2_mark_kernel(const int* __restrict__ vis2,
                                unsigned char* __restrict__ vf2) {
    const int b = blockIdx.y;
    const int i = blockIdx.x * blockDim.x + threadIdx.x;         // 0 .. H*W-1
    const size_t pix = (size_t)b * (HH * WW) + i;
    int v = vis2[pix];
    if (v >= 0) vf2[b * FF + v] = (unsigned char)1;
}

// ---------------------------------------------------------------------------
// Async Global->LDS staging helper (CDNA5 GLOBAL_LOAD_ASYNC_TO_LDS_B64).
// Copies the 8-byte (x,y) prefix of a 12-byte vertex record straight into
// LDS, tracked on ASYNCcnt — no VGPR round trip, one instruction per vertex.
// LDS byte offset = low 32 bits of the generic pointer (ISA LDS aperture:
// flat addr[31:0] is the wave-relative LDS offset).
// ---------------------------------------------------------------------------
__device__ __forceinline__ void async_copy_b64(unsigned lds_off, const void* gaddr) {
#if defined(__gfx1250__)
    asm volatile("global_load_async_to_lds_b64 %0, %1, off"
                 :: "v"(lds_off), "v"(gaddr)
                 : "memory");
#else
    // generic fallback (non-gfx1250 device passes)
    *(float2*)(((char*)(size_t)0) + 0) = *(const float2*)gaddr; (void)lds_off;
#endif
}

__device__ __forceinline__ void wait_asynccnt0() {
#if defined(__gfx1250__)
    asm volatile("s_wait_asynccnt 0x0" ::: "memory");
#endif
}

// ---------------------------------------------------------------------------
// Kernel 3: main flow kernel.
// One block per (batch, 8-row tile). Stages verts_1/verts_2 xy for the whole
// batch into LDS (2 * 6890 float2 = 110,240 B -> 2 blocks per 320KB WGP)
// using async Global->LDS DMA, then walks the full padded 512-wide rows.
// ---------------------------------------------------------------------------
__global__ __launch_bounds__(256) void flow_main_kernel(
    const float*         __restrict__ verts1,
    const int*           __restrict__ vis1,
    const float*         __restrict__ verts2,
    const int*           __restrict__ face,
    const float*         __restrict__ symMask,
    const int*           __restrict__ symtab,
    const unsigned char* __restrict__ vf2,
    float*               __restrict__ outGrid,   // (B,H,WPAD,2)
    float*               __restrict__ outMask)   // (B,H,WPAD)
{
    extern __shared__ float smem[];
    float2* v1s = (float2*)smem;               // VV entries
    float2* v2s = (float2*)(smem + 2 * VV);    // VV entries

    const int b   = blockIdx.y;
    const int tid = threadIdx.x;

    // ---- stage per-batch vertex xy tables into LDS via async DMA ----
    {
        const float* p1 = verts1 + (size_t)b * VV * 3;
        const float* p2 = verts2 + (size_t)b * VV * 3;
        const unsigned l1 = (unsigned)(size_t)(void*)v1s;   // LDS byte offset
        const unsigned l2 = (unsigned)(size_t)(void*)v2s;
        for (int i = tid; i < VV; i += 256) {
            async_copy_b64(l1 + (unsigned)i * 8u, p1 + i * 3);  // x,y of vert i
            async_copy_b64(l2 + (unsigned)i * 8u, p2 + i * 3);
        }
        wait_asynccnt0();          // all async deposits from this wave done
    }
    __syncthreads();               // ...and visible to the whole block

    const int r0 = blockIdx.x * ROWS;
    const unsigned char* vf2b   = vf2 + b * FF;
    const size_t         baseIn = (size_t)b * HH * WW;
    const size_t         baseOut= (size_t)b * HH * WPAD;
    const float          kNorm  = 2.0f / 511.0f;   // Wp-1 == Hp-1 == 511

    for (int idx = tid; idx < ROWS * WPAD; idx += 256) {
        const int X = idx & (WPAD - 1);
        const int y = r0 + (idx >> 9);
        const size_t o = baseOut + (size_t)y * WPAD + X;

        float gx, gy, m;
        if (X < PADX || X >= PADX + WW) {
            // padded border: corr == (0,0) -> normalized (-1,-1), mask 2
            gx = -1.0f; gy = -1.0f; m = 2.0f;
        } else {
            const int    x   = X - PADX;
            const size_t pix = baseIn + (size_t)y * WW + x;

            // prefetch the streaming inputs ahead (gfx1250 global_prefetch_b8)
            __builtin_prefetch(vis1 + pix + 1024, 0, 1);
            __builtin_prefetch(symMask + pix + 1024, 0, 1);

            const int  t  = vis1[pix];
            const bool bg = (t < 0);
            const int  fc = bg ? 0 : t;

            const int i0 = face[fc * 3 + 0];
            const int i1 = face[fc * 3 + 1];
            const int i2 = face[fc * 3 + 2];

            const float2 a  = v1s[i0];
            const float2 bb = v1s[i1];
            const float2 cc = v1s[i2];

            const float e0x = bb.x - a.x, e0y = bb.y - a.y;
            const float e1x = cc.x - a.x, e1y = cc.y - a.y;
            const float e2x = (float)x - a.x, e2y = (float)y - a.y;

            const float d00 = e0x * e0x + e0y * e0y;
            const float d01 = e0x * e1x + e0y * e1y;
            const float d11 = e1x * e1x + e1y * e1y;
            const float d20 = e2x * e0x + e2y * e0y;
            const float d21 = e2x * e1x + e2y * e1y;

            float den = d00 * d11 - d01 * d01;
            den = (fabsf(den) < 1e-12f) ? 1.0f : den;
            const float inv = 1.0f / den;
            const float wv  = (d11 * d20 - d01 * d21) * inv;
            const float ww  = (d00 * d21 - d01 * d20) * inv;
            const float u   = 1.0f - wv - ww;

            const bool vis2f = (vf2b[fc] != 0);

            const int s0 = symtab[i0];
            const int s1 = symtab[i1];
            const int s2 = symtab[i2];
            const float msk = symMask[pix];
            const bool sym_ok  = (s0 >= 0) && (s1 >= 0) && (s2 >= 0) && (msk > 0.0f);
            const bool use_sym = (!vis2f) && sym_ok && (!bg);

            const int t0 = use_sym ? s0 : i0;
            const int t1 = use_sym ? s1 : i1;
            const int t2 = use_sym ? s2 : i2;

            const float2 q0 = v2s[t0];
            const float2 q1 = v2s[t1];
            const float2 q2 = v2s[t2];

            float cx = u * q0.x + wv * q1.x + ww * q2.x;
            float cy = u * q0.y + wv * q1.y + ww * q2.y;

            const bool has = (!bg) && (vis2f || use_sym);
            cx = has ? cx : 0.0f;
            cy = has ? cy : 0.0f;
            cx += (float)PADX;

            m  = bg ? 2.0f : (has ? 0.0f : 1.0f);
            gx = cx * kNorm - 1.0f;
            gy = cy * kNorm - 1.0f;
        }

        ((float2*)outGrid)[o] = make_float2(gx, gy);   // 8B aligned store
        outMask[o] = m;
    }
}

// ---------------------------------------------------------------------------
extern "C" void kernel_launch(void* const* d_in, const int* in_sizes, int n_in,
                              void* d_out, int out_size, void* d_ws, size_t ws_size,
                              hipStream_t stream) {
    (void)in_sizes; (void)n_in; (void)out_size; (void)ws_size;

    const float* verts1  = (const float*)d_in[0];
    const int*   vis1    = (const int*)  d_in[1];
    const float* verts2  = (const float*)d_in[2];
    const int*   vis2    = (const int*)  d_in[3];
    const int*   face    = (const int*)  d_in[4];
    const float* symMask = (const float*)d_in[5];
    const int*   symtab  = (const int*)  d_in[6];

    float* outGrid = (float*)d_out;                                   // B*H*WPAD*2
    float* outMask = (float*)d_out + (size_t)BB * HH * WPAD * 2;      // B*H*WPAD

    unsigned char* vf2 = (unsigned char*)d_ws;                        // B*F bytes

    // 1) zero vf2 (re-done every call: deterministic under graph replay)
    const int n32 = (BB * FF) / 4;                                    // 110208
    vf2_zero_kernel<<<(n32 + 255) / 256, 256, 0, stream>>>((unsigned int*)d_ws, n32);

    // 2) scatter face visibility from view 2
    vf2_mark_kernel<<<dim3((HH * WW) / 256, BB), 256, 0, stream>>>(vis2, vf2);

    // 3) main pass: blocks = (H/ROWS, B), 110,240 B dynamic LDS per block
    const size_t shmem = (size_t)4 * VV * sizeof(float);              // 2 x VV float2
    flow_main_kernel<<<dim3(HH / ROWS, BB), 256, shmem, stream>>>(
        verts1, vis1, verts2, face, symMask, symtab, vf2, outGrid, outMask);
}